// CaFoBlock_5368709120484
// MI455X (gfx1250) — compile-verified
//
#include <hip/hip_runtime.h>

#define D 256
#define LDS_STR 258   // 16-row feature tile stride (floats); 258 % 64 banks = 2 -> conflict-free A reads

typedef __attribute__((ext_vector_type(2))) float v2f;
typedef __attribute__((ext_vector_type(8))) float v8f;

// ---------------------------------------------------------------------------
// Kernel 1: zero the agg accumulator (d_ws is poisoned, must re-zero per call)
// ---------------------------------------------------------------------------
__global__ void zero_f32_kernel(float* __restrict__ p, long long n) {
    long long i = (long long)blockIdx.x * blockDim.x + threadIdx.x;
    long long stride = (long long)gridDim.x * blockDim.x;
    float4* p4 = (float4*)p;
    long long n4 = n >> 2;            // n is a multiple of 4 (N*256)
    for (long long j = i; j < n4; j += stride)
        p4[j] = make_float4(0.f, 0.f, 0.f, 0.f);
}

// ---------------------------------------------------------------------------
// Kernel 2: scatter-add of weighted source rows into destination rows.
// One wave32 per edge; each lane owns two float4 chunks (8 floats) of D=256.
// Edge metadata is forced wave-uniform via readfirstlane -> SMEM s_load path.
// Accumulation uses relaxed agent-scope fetch_add with discarded result ->
// non-returning global_atomic_add_f32 (STOREcnt fire-and-forget).
// Working set (emb 100MB gather + agg 102MB accumulate) is ~L2-resident.
// ---------------------------------------------------------------------------
__global__ void __launch_bounds__(256)
scatter_kernel(const float* __restrict__ emb,
               const int*   __restrict__ src,
               const int*   __restrict__ dst,
               const float* __restrict__ w,
               float* __restrict__ agg,
               int E) {
    int wave = (int)((blockIdx.x * (unsigned)blockDim.x + threadIdx.x) >> 5);
    int lane = threadIdx.x & 31;
    if (wave >= E) return;

    // Provably wave-uniform edge index -> scalar (SMEM) loads of metadata.
    int e = __builtin_amdgcn_readfirstlane(wave);

    int   s  = src[e];
    int   d  = dst[e];
    float wt = w[e];

    const float4* erow = (const float4*)(emb + (long long)s * D);
    float*        arow = agg + (long long)d * D;

    #pragma unroll
    for (int c = 0; c < 2; ++c) {
        int chunk = lane + c * 32;          // 64 float4 chunks cover D=256
        float4 v = erow[chunk];
        float* a = arow + chunk * 4;
        (void)__hip_atomic_fetch_add(a + 0, v.x * wt, __ATOMIC_RELAXED, __HIP_MEMORY_SCOPE_AGENT);
        (void)__hip_atomic_fetch_add(a + 1, v.y * wt, __ATOMIC_RELAXED, __HIP_MEMORY_SCOPE_AGENT);
        (void)__hip_atomic_fetch_add(a + 2, v.z * wt, __ATOMIC_RELAXED, __HIP_MEMORY_SCOPE_AGENT);
        (void)__hip_atomic_fetch_add(a + 3, v.w * wt, __ATOMIC_RELAXED, __HIP_MEMORY_SCOPE_AGENT);
    }
}

// ---------------------------------------------------------------------------
// Kernel 3: gather queried rows + fp32 WMMA GEMM:  out[q,n] = sum_k feats[q,k]*W[n,k] + b[n]
// Block = 256 threads (8 waves), handles 16 query rows x all 256 output cols.
// Each wave computes two 16x16 tiles with V_WMMA_F32_16X16X4_F32 over K=256.
//
// Operand placement per CDNA5 ISA (CDNA5 ISA):
//   A 16x4 f32 : lanes 0-15 row M=lane  {v0:K=0, v1:K=1}; lanes 16-31 row M=lane-16 {v0:K=2, v1:K=3}
//   B 4x16 f32 : lanes 0-15 col N=lane  {v0:K=0, v1:K=1}; lanes 16-31 col N=lane-16 {v0:K=2, v1:K=3}
//   C/D 16x16  : VGPR i, lanes 0-15 -> (M=i,   N=lane); lanes 16-31 -> (M=i+8, N=lane-16)
// ---------------------------------------------------------------------------
__global__ void __launch_bounds__(256)
gather_gemm_kernel(const float* __restrict__ agg,
                   const int*   __restrict__ node_ids,
                   const float* __restrict__ Wm,     // [D x D], row n = output neuron
                   const float* __restrict__ bias,   // [D]
                   float* __restrict__ out,          // [Q x D]
                   int Q) {
    __shared__ float feats[16 * LDS_STR];

    int q0 = blockIdx.x * 16;

    // Gather 16 rows of agg into LDS (coalesced: one full row per iteration)
    for (int r = 0; r < 16; ++r) {
        int q    = q0 + r;
        int node = node_ids[q < Q ? q : (Q - 1)];
        feats[r * LDS_STR + threadIdx.x] = agg[(long long)node * D + threadIdx.x];
    }
    __syncthreads();

    int  waveId = threadIdx.x >> 5;
    int  lane   = threadIdx.x & 31;
    bool hi     = lane >= 16;
    int  m      = lane & 15;
    int  kbase  = hi ? 2 : 0;

    const float* arow = feats + m * LDS_STR + kbase;   // A: row m, K-pair select by half-wave

    #pragma unroll
    for (int t = 0; t < 2; ++t) {
        int n0 = (waveId * 2 + t) * 16;
        int n  = n0 + m;
        const float* brow = Wm + (long long)n * D + kbase;  // B[k][n] = W[n][k]

        v8f acc = {};
        #pragma unroll 8
        for (int k = 0; k < D; k += 4) {
            v2f a;  a.x  = arow[k];  a.y  = arow[k + 1];
            v2f bv; bv.x = brow[k];  bv.y = brow[k + 1];
            acc = __builtin_amdgcn_wmma_f32_16x16x4_f32(
                false, a, false, bv, (short)0, acc, false, false);
        }

        float bn = bias[n];
        #pragma unroll
        for (int i = 0; i < 8; ++i) {
            int row = (hi ? 8 : 0) + i;
            int q   = q0 + row;
            if (q < Q)
                out[(long long)q * D + n0 + m] = acc[i] + bn;
        }
    }
}

// ---------------------------------------------------------------------------
extern "C" void kernel_launch(void* const* d_in, const int* in_sizes, int n_in,
                              void* d_out, int out_size, void* d_ws, size_t ws_size,
                              hipStream_t stream) {
    const float* emb  = (const float*)d_in[0];
    const int*   esrc = (const int*)  d_in[1];
    const int*   edst = (const int*)  d_in[2];
    const float* ew   = (const float*)d_in[3];
    const int*   nid  = (const int*)  d_in[4];
    const float* Wm   = (const float*)d_in[5];
    const float* bias = (const float*)d_in[6];

    int N = in_sizes[0] / D;
    int E = in_sizes[1];
    int Q = in_sizes[4];

    float* agg = (float*)d_ws;                 // [N x D] accumulator
    long long aggN = (long long)N * D;

    zero_f32_kernel<<<2048, 256, 0, stream>>>(agg, aggN);

    long long scatterThreads = (long long)E * 32;      // one wave per edge
    int sblocks = (int)((scatterThreads + 255) / 256);
    scatter_kernel<<<sblocks, 256, 0, stream>>>(emb, esrc, edst, ew, agg, E);

    int gblocks = (Q + 15) / 16;
    gather_gemm_kernel<<<gblocks, 256, 0, stream>>>(agg, nid, Wm, bias,
                                                    (float*)d_out, Q);
}